// NGCF_86938728006178
// MI455X (gfx1250) — compile-verified
//
#include <hip/hip_runtime.h>

// ---------------- problem constants ----------------
#define N_USERS 100000
#define N_ITEMS 50000
#define NTOT    150000
#define NPAD    150016   // round up to multiple of 128 (8 waves x 16 rows)
#define HOPS_   3
#define NNZ_    2400000
#define B_      4096
#define K_      4
#define DECAY_  1e-4f
#define EPS_    1e-12f

typedef __attribute__((ext_vector_type(2))) float v2f;
typedef __attribute__((ext_vector_type(4))) float v4f;
typedef __attribute__((ext_vector_type(8))) float v8f;

__device__ inline v8f splat8(float x) {
  v8f r;
#pragma unroll
  for (int i = 0; i < 8; ++i) r[i] = x;
  return r;
}

__device__ inline v8f wmma4(v2f a, v2f b, v8f c) {
  // D = A(16x4,f32) x B(4x16,f32) + C(16x16,f32)
  return __builtin_amdgcn_wmma_f32_16x16x4_f32(
      /*neg_a=*/false, a, /*neg_b=*/false, b,
      /*c_mod=*/(short)0, c, /*reuse_a=*/false, /*reuse_b=*/false);
}

// ---------------- ego = concat(user,item); layer0 copy ----------------
__global__ __launch_bounds__(256) void init_ego(const float* __restrict__ ue,
                                                const float* __restrict__ ie,
                                                float* __restrict__ ego,
                                                float* __restrict__ all_emb) {
  int i = blockIdx.x * blockDim.x + threadIdx.x;   // NPAD*16 float4 slots
  if (i >= NPAD * 16) return;
  int node = i >> 4;
  int d4 = (i & 15) * 4;
  float4 v = make_float4(0.f, 0.f, 0.f, 0.f);
  if (node < N_USERS)   v = *(const float4*)(ue + (size_t)node * 64 + d4);
  else if (node < NTOT) v = *(const float4*)(ie + (size_t)(node - N_USERS) * 64 + d4);
  *(float4*)(ego + (size_t)node * 64 + d4) = v;
  *(float4*)(all_emb + (size_t)node * 256 + d4) = v;   // layer 0
}

// ---------------- side = segment_sum(vals * ego[cols], rows) ----------------
__global__ __launch_bounds__(256) void spmm(const float* __restrict__ ego,
                                            const float* __restrict__ vals,
                                            const int* __restrict__ rows,
                                            const int* __restrict__ cols,
                                            float* __restrict__ side) {
  long long t = (long long)blockIdx.x * blockDim.x + threadIdx.x;
  int e = (int)(t >> 3);
  if (e >= NNZ_) return;
  int q = (int)(t & 7);                 // 8 floats per thread
  int r = rows[e];
  int c = cols[e];
  float v = vals[e];
  const float* src = ego + (size_t)c * 64 + q * 8;
  float4 a = *(const float4*)(src);
  float4 b = *(const float4*)(src + 4);
  float* dst = side + (size_t)r * 64 + q * 8;
  atomicAdd(dst + 0, v * a.x);
  atomicAdd(dst + 1, v * a.y);
  atomicAdd(dst + 2, v * a.z);
  atomicAdd(dst + 3, v * a.w);
  atomicAdd(dst + 4, v * b.x);
  atomicAdd(dst + 5, v * b.y);
  atomicAdd(dst + 6, v * b.z);
  atomicAdd(dst + 7, v * b.w);
}

// ---------------- fused: leaky(side@Wgc + b + (ego*side)@Wbi + b) ----------------
// writes ego (unnormalized, in place) and all_emb[:,layer,:] (row-l2-normalized)
// LDS weight layout: sW[(kpair*64 + col)*4 + {0,1,2,3}] =
//   {Wgc[2p][col], Wgc[2p+1][col], Wbi[2p][col], Wbi[2p+1][col]}
// -> one ds_load_b128 yields both B fragments (K-pair for Wgc and Wbi) already
//    packed into even-aligned v2f operand pairs; zero packing movs.
__global__ __launch_bounds__(256) void hop_gemm(
    const float* __restrict__ Wgc, const float* __restrict__ bgc,
    const float* __restrict__ Wbi, const float* __restrict__ bbi,
    const float* __restrict__ side, float* __restrict__ ego,
    float* __restrict__ all_emb, int layer) {
  __shared__ float sW[32 * 64 * 4];     // 32 KB
  int tid = threadIdx.x;
  for (int i = tid; i < 4096; i += 256) {
    int row = i >> 6, col = i & 63;
    int p = row >> 1, h = row & 1;
    float* dptr = &sW[(((p << 6) + col) << 2) + h];
    dptr[0] = Wgc[i];
    dptr[2] = Wbi[i];
  }

  int wave = tid >> 5, lane = tid & 31;
  int c = lane & 15;        // output column within 16-wide tile / A row (M)
  int g = lane >> 4;        // half-wave selector (K-pair for A/B, +8 rows for D)
  size_t row0 = (size_t)blockIdx.x * 128 + (size_t)wave * 16;

  // accumulators seeded with fused bias (same per output column)
  v8f acc0 = splat8(bgc[c]      + bbi[c]);
  v8f acc1 = splat8(bgc[16 + c] + bbi[16 + c]);
  v8f acc2 = splat8(bgc[32 + c] + bbi[32 + c]);
  v8f acc3 = splat8(bgc[48 + c] + bbi[48 + c]);

  const float* srow = side + row0 * 64;
  const float* erow = ego + row0 * 64;
  const int aoff = c * 64 + 2 * g;      // A fragment: row c, K-pair base 2g

  __syncthreads();                      // sW ready

#pragma unroll
  for (int kb = 0; kb < 16; ++kb) {
    const int kk = kb * 4;
    v2f s = *(const v2f*)(srow + aoff + kk);   // side fragment
    v2f e = *(const v2f*)(erow + aoff + kk);   // ego fragment
    v2f pr = s * e;                            // (ego .* side) fragment
    // weight quad base for K-pair (2*kb + g), column c (tile t at +64*t floats)
    const float* wrow = &sW[((((2 * kb + g) << 6) + c) << 2)];
    v4f w0 = *(const v4f*)(wrow);
    v4f w1 = *(const v4f*)(wrow + 64);
    v4f w2 = *(const v4f*)(wrow + 128);
    v4f w3 = *(const v4f*)(wrow + 192);
    v2f b0 = {w0.x, w0.y}, c0 = {w0.z, w0.w};
    v2f b1 = {w1.x, w1.y}, c1 = {w1.z, w1.w};
    v2f b2 = {w2.x, w2.y}, c2 = {w2.z, w2.w};
    v2f b3 = {w3.x, w3.y}, c3 = {w3.z, w3.w};
    acc0 = wmma4(s, b0, acc0); acc0 = wmma4(pr, c0, acc0);
    acc1 = wmma4(s, b1, acc1); acc1 = wmma4(pr, c1, acc1);
    acc2 = wmma4(s, b2, acc2); acc2 = wmma4(pr, c2, acc2);
    acc3 = wmma4(s, b3, acc3); acc3 = wmma4(pr, c3, acc3);
  }

  // leaky relu (slope 0.2)
#pragma unroll
  for (int v = 0; v < 8; ++v) {
    acc0[v] = acc0[v] > 0.f ? acc0[v] : 0.2f * acc0[v];
    acc1[v] = acc1[v] > 0.f ? acc1[v] : 0.2f * acc1[v];
    acc2[v] = acc2[v] > 0.f ? acc2[v] : 0.2f * acc2[v];
    acc3[v] = acc3[v] > 0.f ? acc3[v] : 0.2f * acc3[v];
  }

  // per-row sum of squares: row (v + 8g) lives in 16 lanes (c) x 4 tiles
  float ss[8];
#pragma unroll
  for (int v = 0; v < 8; ++v)
    ss[v] = acc0[v] * acc0[v] + acc1[v] * acc1[v] +
            acc2[v] * acc2[v] + acc3[v] * acc3[v];
#pragma unroll
  for (int m = 1; m <= 8; m <<= 1) {
#pragma unroll
    for (int v = 0; v < 8; ++v) ss[v] += __shfl_xor(ss[v], m, 32);
  }
  float inv[8];
#pragma unroll
  for (int v = 0; v < 8; ++v) inv[v] = 1.f / fmaxf(sqrtf(ss[v]), EPS_);

#pragma unroll
  for (int v = 0; v < 8; ++v) {
    size_t r = row0 + v + 8 * g;
    float* eo = ego + r * 64 + c;
    float* ao = all_emb + r * 256 + (size_t)layer * 64 + c;
    eo[0]  = acc0[v]; eo[16] = acc1[v]; eo[32] = acc2[v]; eo[48] = acc3[v];
    ao[0]  = acc0[v] * inv[v];
    ao[16] = acc1[v] * inv[v];
    ao[32] = acc2[v] * inv[v];
    ao[48] = acc3[v] * inv[v];
  }
}

// ---------------- loss: one 64-thread block per sample ----------------
__global__ __launch_bounds__(64) void loss_kernel(const float* __restrict__ all_emb,
                                                  const int* __restrict__ users,
                                                  const int* __restrict__ pos,
                                                  const int* __restrict__ neg,
                                                  float* __restrict__ accum) {
  __shared__ float sred[64];
  int b = blockIdx.x, d = threadIdx.x;

  auto red = [&](float v) -> float {
    sred[d] = v; __syncthreads();
    if (d < 32) sred[d] += sred[d + 32]; __syncthreads();
    if (d < 16) sred[d] += sred[d + 16]; __syncthreads();
    if (d < 8)  sred[d] += sred[d + 8];  __syncthreads();
    if (d < 4)  sred[d] += sred[d + 4];  __syncthreads();
    if (d < 2)  sred[d] += sred[d + 2];  __syncthreads();
    if (d < 1)  sred[d] += sred[d + 1];  __syncthreads();
    float r = sred[0]; __syncthreads();
    return r;
  };

  int u = users[b], p = pos[b];
  const float* ua = all_emb + (size_t)u * 256;
  const float* pa = all_emb + (size_t)(N_USERS + p) * 256;
  float u0 = ua[d];
  float ue = 0.25f * (u0 + ua[64 + d] + ua[128 + d] + ua[192 + d]);
  float p0 = pa[d];
  float pe = 0.25f * (p0 + pa[64 + d] + pa[128 + d] + pa[192 + d]);
  float ps = red(ue * pe);
  float regp = u0 * u0 + p0 * p0;
  float se = 0.f;
  for (int k = 0; k < K_; ++k) {
    int n = neg[b * K_ + k];
    const float* na = all_emb + (size_t)(N_USERS + n) * 256;
    float n0 = na[d];
    float ne = 0.25f * (n0 + na[64 + d] + na[128 + d] + na[192 + d]);
    float ns = red(ue * ne);
    regp += n0 * n0;
    se += expf(ns - ps);
  }
  float reg = red(regp);
  if (d == 0) {
    atomicAdd(&accum[0], log1pf(se));
    atomicAdd(&accum[1], 0.5f * reg);
  }
}

__global__ void finalize(const float* __restrict__ accum, float* __restrict__ out) {
  if (threadIdx.x == 0 && blockIdx.x == 0) {
    float mf = accum[0] * (1.0f / B_);
    float emb = DECAY_ * accum[1] * (1.0f / B_);
    out[0] = mf + emb;
    out[1] = mf;
    out[2] = emb;
  }
}

// ---------------- host launch ----------------
extern "C" void kernel_launch(void* const* d_in, const int* in_sizes, int n_in,
                              void* d_out, int out_size, void* d_ws, size_t ws_size,
                              hipStream_t stream) {
  const float* user_emb = (const float*)d_in[0];
  const float* item_emb = (const float*)d_in[1];
  const float* adj_vals = (const float*)d_in[2];
  const float* W_gc = (const float*)d_in[3];
  const float* b_gc = (const float*)d_in[4];
  const float* W_bi = (const float*)d_in[5];
  const float* b_bi = (const float*)d_in[6];
  const int* adj_rows = (const int*)d_in[7];
  const int* adj_cols = (const int*)d_in[8];
  const int* users = (const int*)d_in[9];
  const int* pos_items = (const int*)d_in[10];
  const int* neg_items = (const int*)d_in[11];

  float* ego = (float*)d_ws;                       // NPAD*64
  float* side = ego + (size_t)NPAD * 64;           // NPAD*64
  float* all_emb = side + (size_t)NPAD * 64;       // NPAD*4*64
  float* accum = all_emb + (size_t)NPAD * 256;     // 2 floats

  init_ego<<<(NPAD * 16 + 255) / 256, 256, 0, stream>>>(user_emb, item_emb, ego, all_emb);

  for (int k = 0; k < HOPS_; ++k) {
    hipMemsetAsync(side, 0, (size_t)NPAD * 64 * sizeof(float), stream);
    spmm<<<(int)(((long long)NNZ_ * 8 + 255) / 256), 256, 0, stream>>>(
        ego, adj_vals, adj_rows, adj_cols, side);
    hop_gemm<<<NPAD / 128, 256, 0, stream>>>(
        W_gc + (size_t)k * 4096, b_gc + (size_t)k * 64,
        W_bi + (size_t)k * 4096, b_bi + (size_t)k * 64,
        side, ego, all_emb, k + 1);
  }

  hipMemsetAsync(accum, 0, 2 * sizeof(float), stream);
  loss_kernel<<<B_, 64, 0, stream>>>(all_emb, users, pos_items, neg_items, accum);
  finalize<<<1, 1, 0, stream>>>(accum, (float*)d_out);
}